// ImprovedGATv2_53463752900665
// MI455X (gfx1250) — compile-verified
//
#include <hip/hip_runtime.h>
#include <hip/hip_bf16.h>
#include <math.h>

typedef __attribute__((ext_vector_type(2))) float v2f;
typedef __attribute__((ext_vector_type(8))) float v8f;

#define HIDC 64

__device__ __forceinline__ float elu1(float x) { return x > 0.f ? x : (expf(x) - 1.f); }
__device__ __forceinline__ float lrelu(float x) { return x > 0.f ? x : 0.2f * x; }

// order-preserving float <-> uint encoding for atomicMax on floats
__device__ __forceinline__ unsigned fenc(float f) {
    unsigned b = __float_as_uint(f);
    return (b & 0x80000000u) ? ~b : (b | 0x80000000u);
}
__device__ __forceinline__ float fdec(unsigned u) {
    unsigned b = (u & 0x80000000u) ? (u & 0x7FFFFFFFu) : ~u;
    return __uint_as_float(b);
}

// ---------------------------------------------------------------------------
// WMMA f32 GEMM:  C[M,Nout] = act(A[M,K] @ W[K,Nout] + bias)
// One wave computes a 16 x (16*NT) strip: NT accumulators share one A frag
// per K-step -> NT v_wmma_f32_16x16x4_f32 per A load. Branch-free inner loop
// with running pointers (no per-iteration 64-bit multiplies).
// Out-of-range columns: B load address clamped (in-bounds; garbage only lands
// in accumulator columns that are never stored); store is guarded.
// A frag (16x4): lane<16 -> (K0,K1) of row m0+lane ; lane>=16 -> (K2,K3)
// B frag (4x16): lane<16 -> (K0,K1) of col n0+lane ; lane>=16 -> (K2,K3)
// C/D tile     : vgpr v, lane l -> row m0 + v + 8*(l>=16), col n0 + (l&15)
// M multiple of 16; K multiple of 4 (holds for all shapes here).
// ---------------------------------------------------------------------------
template <int NT>
__global__ void gemm_wmma_f32(const float* __restrict__ A, const float* __restrict__ W,
                              const float* __restrict__ bias, float* __restrict__ C,
                              int M, int K, int Nout, int act) {
    const int lane = threadIdx.x;              // 0..31
    const int wave = threadIdx.y;              // 0..3
    const int m0 = (blockIdx.y * 4 + wave) * 16;
    const int n0 = blockIdx.x * (16 * NT);
    if (m0 >= M) return;                       // wave-uniform exit (EXEC all-1 for WMMA)

    const int half = lane >> 4;                // 0 or 1
    const int lr   = lane & 15;

    const float* ap = A + (size_t)(m0 + lr) * K + 2 * half;

    int colv[NT];
    const float* wp[NT];
    v8f acc[NT];
    const v8f vzero = {};
#pragma unroll
    for (int j = 0; j < NT; ++j) {
        int c = n0 + j * 16 + lr;
        colv[j] = c;
        int cl = (c < Nout) ? c : (Nout - 1);  // clamped, always in-bounds
        wp[j] = W + cl + (size_t)(2 * half) * Nout;
        acc[j] = vzero;
    }
    const int wstep = 4 * Nout;                // floats advanced per K-step

    for (int k = 0; k < K; k += 4) {
        v2f a;
        a.x = ap[0];
        a.y = ap[1];
        ap += 4;
#pragma unroll
        for (int j = 0; j < NT; ++j) {
            v2f b;
            b.x = wp[j][0];
            b.y = wp[j][Nout];
            wp[j] += wstep;
            acc[j] = __builtin_amdgcn_wmma_f32_16x16x4_f32(false, a, false, b,
                                                           (short)0, acc[j], false, false);
        }
    }

#pragma unroll
    for (int j = 0; j < NT; ++j) {
        int col = colv[j];
        if (col < Nout) {
            const float bv = bias[col];
#pragma unroll
            for (int v = 0; v < 8; ++v) {
                int row = m0 + v + half * 8;
                float val = acc[j][v] + bv;
                if (act == 1) val = elu1(val);
                C[(size_t)row * Nout + col] = val;
            }
        }
    }
}

// ---------------------------------------------------------------------------
// Per-layer segment-state init: AGG=0, SUM=0, MAXENC=encode(-inf)
// ---------------------------------------------------------------------------
__global__ void init_seg(float* __restrict__ AGG, float* __restrict__ SUM,
                         unsigned* __restrict__ MAXE, int nAgg, int nSeg) {
    int t = blockIdx.x * blockDim.x + threadIdx.x;
    if (t < nAgg) AGG[t] = 0.f;
    if (t < nSeg) { SUM[t] = 0.f; MAXE[t] = 0x007FFFFFu; }  // encode(-inf)
}

// ---------------------------------------------------------------------------
// Pass 1: per-(edge,head) attention logit + segment max (atomicMax, encoded)
// Edge list = edge_index columns (e<E) followed by N self loops.
// ---------------------------------------------------------------------------
__global__ void edge_logits(const float* __restrict__ XL, const float* __restrict__ XR,
                            const int* __restrict__ ei, const float* __restrict__ att,
                            unsigned* __restrict__ MAXE, float* __restrict__ LOG,
                            int E, int Nn, int H) {
    int t = blockIdx.x * blockDim.x + threadIdx.x;
    int EE = E + Nn;
    if (t >= EE * H) return;
    int e = t / H, h = t - e * H;
    int s = (e < E) ? ei[e] : (e - E);
    int d = (e < E) ? ei[E + e] : (e - E);

    const float4* xj = (const float4*)(XL + ((size_t)s * H + h) * HIDC);
    const float4* xi = (const float4*)(XR + ((size_t)d * H + h) * HIDC);
    const float4* at = (const float4*)(att + (size_t)h * HIDC);

    float acc = 0.f;
#pragma unroll 4
    for (int c = 0; c < HIDC / 4; ++c) {
        float4 a = xj[c], b = xi[c], w = at[c];
        acc += lrelu(a.x + b.x) * w.x;
        acc += lrelu(a.y + b.y) * w.y;
        acc += lrelu(a.z + b.z) * w.z;
        acc += lrelu(a.w + b.w) * w.w;
    }
    LOG[t] = acc;
    atomicMax(MAXE + (size_t)d * H + h, fenc(acc));
}

// ---------------------------------------------------------------------------
// Pass 2: p = exp(logit - segmax); segment sum (atomicAdd). LOG := p in place.
// ---------------------------------------------------------------------------
__global__ void edge_expsum(const int* __restrict__ ei, const unsigned* __restrict__ MAXE,
                            float* __restrict__ LOG, float* __restrict__ SUM,
                            int E, int Nn, int H) {
    int t = blockIdx.x * blockDim.x + threadIdx.x;
    int EE = E + Nn;
    if (t >= EE * H) return;
    int e = t / H, h = t - e * H;
    int d = (e < E) ? ei[E + e] : (e - E);
    float m = fdec(MAXE[(size_t)d * H + h]);
    float p = expf(LOG[t] - m);
    LOG[t] = p;
    atomicAdd(SUM + (size_t)d * H + h, p);
}

// ---------------------------------------------------------------------------
// Pass 3: AGG[dst, h*64+c] += alpha * xj[c].  16 lanes per (edge,head),
// float4 gather per lane -> coalesced 256B per message, f32 atomics to L2.
// ---------------------------------------------------------------------------
__global__ void edge_scatter(const float* __restrict__ XL, const int* __restrict__ ei,
                             const float* __restrict__ LOG, const float* __restrict__ SUM,
                             float* __restrict__ AGG, int E, int Nn, int H) {
    int t = blockIdx.x * blockDim.x + threadIdx.x;
    int EE = E + Nn;
    if (t >= EE * H * 16) return;
    int c4 = t & 15;
    int eh = t >> 4;
    int e = eh / H, h = eh - e * H;
    int s = (e < E) ? ei[e] : (e - E);
    int d = (e < E) ? ei[E + e] : (e - E);

    float alpha = LOG[eh] / SUM[(size_t)d * H + h];
    float4 xj = ((const float4*)(XL + ((size_t)s * H + h) * HIDC))[c4];
    float* o = AGG + ((size_t)d * H + h) * HIDC + c4 * 4;
    atomicAdd(o + 0, alpha * xj.x);
    atomicAdd(o + 1, alpha * xj.y);
    atomicAdd(o + 2, alpha * xj.z);
    atomicAdd(o + 3, alpha * xj.w);
}

// ---------------------------------------------------------------------------
// Finalize: h = LN(AGG + layer_bias + SKIP) [* elu].  One wave32 per node,
// D/32 channels per lane, reductions via __shfl_xor (wave32 native).
// ---------------------------------------------------------------------------
__global__ void finalize_ln(const float* __restrict__ AGG, const float* __restrict__ SK,
                            const float* __restrict__ lbias, const float* __restrict__ gamma,
                            const float* __restrict__ beta, float* __restrict__ hout,
                            int Nn, int D, int act) {
    int lane = threadIdx.x;                    // 0..31
    int n = blockIdx.x * blockDim.y + threadIdx.y;
    if (n >= Nn) return;
    int nch = D / 32;                          // 8 (D=256) or 2 (D=64)
    float v[8];
    float sum = 0.f;
    for (int j = 0; j < nch; ++j) {
        int idx = j * 32 + lane;
        v[j] = AGG[(size_t)n * D + idx] + lbias[idx] + SK[(size_t)n * D + idx];
        sum += v[j];
    }
#pragma unroll
    for (int o = 16; o >= 1; o >>= 1) sum += __shfl_xor(sum, o, 32);
    float mu = sum / (float)D;
    float var = 0.f;
    for (int j = 0; j < nch; ++j) { float dd = v[j] - mu; var += dd * dd; }
#pragma unroll
    for (int o = 16; o >= 1; o >>= 1) var += __shfl_xor(var, o, 32);
    float rstd = rsqrtf(var / (float)D + 1e-5f);
    for (int j = 0; j < nch; ++j) {
        int idx = j * 32 + lane;
        float o_ = (v[j] - mu) * rstd * gamma[idx] + beta[idx];
        if (act) o_ = elu1(o_);
        hout[(size_t)n * D + idx] = o_;
    }
}

// ---------------------------------------------------------------------------
extern "C" void kernel_launch(void* const* d_in, const int* in_sizes, int n_in,
                              void* d_out, int out_size, void* d_ws, size_t ws_size,
                              hipStream_t stream) {
    const int Nn = in_sizes[0] / 300;          // 30000
    const int E  = in_sizes[1] / 2;            // 480000
    const int EE = E + Nn;

    const float* x    = (const float*)d_in[0];
    const int*   ei   = (const int*)d_in[1];
    const float* W_in = (const float*)d_in[2];
    const float* b_in = (const float*)d_in[3];
    const float* W1 = (const float*)d_in[34];
    const float* b1 = (const float*)d_in[35];
    const float* W2 = (const float*)d_in[36];
    const float* b2 = (const float*)d_in[37];

    // workspace carve-up (floats)
    float* p = (float*)d_ws;
    const size_t bigN = (size_t)Nn * 256;
    float* hA  = p; p += bigN;
    float* hB  = p; p += bigN;
    float* XL  = p; p += bigN;
    float* XR  = p; p += bigN;
    float* SK  = p; p += bigN;
    float* AGG = p; p += bigN;
    float* LOG = p; p += (size_t)EE * 4;
    float* SUM = p; p += (size_t)Nn * 4;
    unsigned* MAXE = (unsigned*)p;

    const dim3 gblk(32, 4);
    const int mtiles = (Nn + 63) / 64;

    // h0 = elu(x @ W_in + b_in)   [N,64] : one 16x64 strip covers all cols
    gemm_wmma_f32<4><<<dim3(1, mtiles), gblk, 0, stream>>>(x, W_in, b_in, hA,
                                                           Nn, 300, 64, 1);

    float* h_in = hA;
    float* h_out = hB;
    const int HEADS[3]  = {4, 4, 1};
    const int DIN[3]    = {64, 256, 256};
    const int DOUT[3]   = {256, 256, 64};      // concat,concat,mean(H=1)

    for (int i = 0; i < 3; ++i) {
        const int base = 4 + 10 * i;
        const float* Wl  = (const float*)d_in[base + 0];
        const float* bl  = (const float*)d_in[base + 1];
        const float* Wr  = (const float*)d_in[base + 2];
        const float* br  = (const float*)d_in[base + 3];
        const float* att = (const float*)d_in[base + 4];
        const float* lb  = (const float*)d_in[base + 5];
        const float* Wsk = (const float*)d_in[base + 6];
        const float* bsk = (const float*)d_in[base + 7];
        const float* gam = (const float*)d_in[base + 8];
        const float* bet = (const float*)d_in[base + 9];
        const int H = HEADS[i], Din = DIN[i], D = DOUT[i];
        const int HW = H * HIDC;               // xl/xr width (256 or 64)

        gemm_wmma_f32<4><<<dim3(HW / 64, mtiles), gblk, 0, stream>>>(h_in, Wl, bl, XL,
                                                                     Nn, Din, HW, 0);
        gemm_wmma_f32<4><<<dim3(HW / 64, mtiles), gblk, 0, stream>>>(h_in, Wr, br, XR,
                                                                     Nn, Din, HW, 0);
        gemm_wmma_f32<4><<<dim3(D / 64, mtiles), gblk, 0, stream>>>(h_in, Wsk, bsk, SK,
                                                                    Nn, Din, D, 0);

        int nAgg = Nn * D, nSeg = Nn * H;
        init_seg<<<(nAgg + 255) / 256, 256, 0, stream>>>(AGG, SUM, MAXE, nAgg, nSeg);

        int nEH = EE * H;
        edge_logits<<<(nEH + 255) / 256, 256, 0, stream>>>(XL, XR, ei, att, MAXE, LOG,
                                                           E, Nn, H);
        edge_expsum<<<(nEH + 255) / 256, 256, 0, stream>>>(ei, MAXE, LOG, SUM, E, Nn, H);
        int nSc = nEH * 16;
        edge_scatter<<<(nSc + 255) / 256, 256, 0, stream>>>(XL, ei, LOG, SUM, AGG,
                                                            E, Nn, H);

        finalize_ln<<<(Nn + 7) / 8, dim3(32, 8), 0, stream>>>(AGG, SK, lb, gam, bet,
                                                              h_out, Nn, D, (i < 2) ? 1 : 0);
        float* tmp = h_in; h_in = h_out; h_out = tmp;
    }

    // MLP head: t1 = elu(h @ W1 + b1) [N,32];  out = t1 @ W2 + b2 [N,8]
    float* T1 = XL;  // reuse
    gemm_wmma_f32<2><<<dim3(1, mtiles), gblk, 0, stream>>>(h_in, W1, b1, T1,
                                                           Nn, 64, 32, 1);
    gemm_wmma_f32<1><<<dim3(1, mtiles), gblk, 0, stream>>>(T1, W2, b2, (float*)d_out,
                                                           Nn, 32, 8, 0);
}